// LNN_82867099009185
// MI455X (gfx1250) — compile-verified
//
#include <hip/hip_runtime.h>
#include <math.h>

#define H_DIM  500
#define HP     512         // padded LDS row stride (per channel)
#define ZDIM   4
#define CH     16          // 1 value + 4 grad + 10 hess(upper) + 1 pad
#define NCH    15
#define DT_F   0.01f
#define BATCH  8192
#define SPB    4           // samples per block (shares A fragments 4x)
#define SBUF   (CH * HP)   // floats per sample jet buffer

typedef __attribute__((ext_vector_type(2))) float v2f;
typedef __attribute__((ext_vector_type(4))) float v4f;
typedef __attribute__((ext_vector_type(8))) float v8f;

__device__ __forceinline__ float softplus_f(float u) {
    return fmaxf(u, 0.0f) + log1pf(__expf(-fabsf(u)));
}
__device__ __forceinline__ float sigmoid_f(float u) {
    return 1.0f / (1.0f + __expf(-u));
}

// C[s] = W(500x500) @ state[s](500 x 16ch) for 4 samples, fp32 WMMA 16x16x4.
// Each wave owns 4 row tiles, processed as 2 pairs; per k-step we load 2 A
// fragments (global/L2) and 4 B fragments (LDS) and issue 8 independent WMMAs
// -> A traffic amortized 4x across samples, 8-deep WMMA ILP.
__device__ __forceinline__ void layer_gemm_wmma(const float* __restrict__ W,
                                                const float* inBase, float* outBase,
                                                int lane, int wave) {
    const int ch    = lane & 15;          // N index (jet channel)
    const int mrow  = lane & 15;          // M index within tile for A
    const int khalf = (lane >> 4) << 1;   // lanes 16-31 hold K+2,K+3

#pragma unroll
    for (int rt = 0; rt < 2; ++rt) {
        const int mt0 = wave * 4 + rt * 2;             // tiles mt0, mt0+1
        const int r0  = mt0 * 16 + mrow;
        const int r1  = r0 + 16;
        const int c0  = r0 < H_DIM ? r0 : (H_DIM - 1); // clamp pad rows
        const int c1  = r1 < H_DIM ? r1 : (H_DIM - 1);
        const float* wrow0 = W + c0 * H_DIM;
        const float* wrow1 = W + c1 * H_DIM;

        v8f acc[2 * SPB];
#pragma unroll
        for (int i = 0; i < 2 * SPB; ++i) acc[i] = (v8f){};

        for (int k = 0; k < H_DIM; k += 4) {
            v2f a0 = *(const v2f*)(wrow0 + k + khalf);
            v2f a1 = *(const v2f*)(wrow1 + k + khalf);
#pragma unroll
            for (int si = 0; si < SPB; ++si) {
                v2f b = *(const v2f*)(inBase + si * SBUF + ch * HP + k + khalf);
                acc[si]       = __builtin_amdgcn_wmma_f32_16x16x4_f32(
                    false, a0, false, b, (short)0, acc[si],       false, false);
                acc[SPB + si] = __builtin_amdgcn_wmma_f32_16x16x4_f32(
                    false, a1, false, b, (short)0, acc[SPB + si], false, false);
            }
        }
        // D layout: element r -> C[M = mt*16 + r + 8*(lane>=16)][N = lane&15].
        // Unconditional 16B stores into padded rows (32B-aligned -> b128).
        const int mb0 = mt0 * 16 + ((lane >> 4) << 3);
#pragma unroll
        for (int si = 0; si < SPB; ++si) {
            float* out0 = outBase + si * SBUF + ch * HP + mb0;
            v8f a = acc[si], b = acc[SPB + si];
            *(v4f*)(out0)      = (v4f){a[0], a[1], a[2], a[3]};
            *(v4f*)(out0 + 4)  = (v4f){a[4], a[5], a[6], a[7]};
            *(v4f*)(out0 + 16) = (v4f){b[0], b[1], b[2], b[3]};
            *(v4f*)(out0 + 20) = (v4f){b[4], b[5], b[6], b[7]};
        }
    }
}

// Pointwise softplus jet update on pre-activations (adds bias to value ch).
__device__ __forceinline__ void jet_activation(float* base, const float* __restrict__ bias,
                                               int tid) {
#pragma unroll
    for (int si = 0; si < SPB; ++si) {
        float* buf = base + si * SBUF;
        for (int n = tid; n < H_DIM; n += 256) {
            float p[NCH];
#pragma unroll
            for (int c = 0; c < NCH; ++c) p[c] = buf[c * HP + n];
            const float u  = p[0] + bias[n];
            const float sg = sigmoid_f(u);
            const float sd = sg * (1.0f - sg);
            float g[4];
#pragma unroll
            for (int i = 0; i < 4; ++i) g[i] = p[1 + i];
            buf[0 * HP + n] = softplus_f(u);
#pragma unroll
            for (int i = 0; i < 4; ++i) buf[(1 + i) * HP + n] = sg * g[i];
            int c = 5;
#pragma unroll
            for (int i = 0; i < 4; ++i)
#pragma unroll
                for (int j = i; j < 4; ++j, ++c)
                    buf[c * HP + n] = sd * g[i] * g[j] + sg * p[c];
            buf[15 * HP + n] = 0.0f;
        }
    }
}

__global__ __launch_bounds__(256)
void lnn_f_kernel(const float* __restrict__ z_in,
                  const float* __restrict__ W1, const float* __restrict__ b1,
                  const float* __restrict__ W2, const float* __restrict__ b2,
                  const float* __restrict__ W3, const float* __restrict__ b3,
                  const float* __restrict__ W4, const float* __restrict__ b4,
                  float* __restrict__ f_out) {
    extern __shared__ float smem[];
    float* bufA = smem;                    // SPB * 16 * 512
    float* bufB = smem + SPB * SBUF;       // SPB * 16 * 512
    float* red  = smem + 2 * SPB * SBUF;   // 256 partials + 64 finals
    float* zsh  = red + 320;               // SPB * 4 inputs

    const int tid  = threadIdx.x;
    const int lane = tid & 31;
    const int wave = tid >> 5;

    if (tid < SPB * ZDIM) zsh[tid] = z_in[blockIdx.x * (SPB * ZDIM) + tid];
    __syncthreads();

    // ---- layer 1 (500x4): jet init + softplus jet, scalar per thread ----
#pragma unroll
    for (int si = 0; si < SPB; ++si) {
        const float z0 = zsh[si * 4 + 0], z1 = zsh[si * 4 + 1];
        const float z2 = zsh[si * 4 + 2], z3 = zsh[si * 4 + 3];
        float* buf = bufA + si * SBUF;
        for (int n = tid; n < H_DIM; n += 256) {
            float g[4];
#pragma unroll
            for (int i = 0; i < 4; ++i) g[i] = W1[n * 4 + i];
            const float u  = b1[n] + g[0] * z0 + g[1] * z1 + g[2] * z2 + g[3] * z3;
            const float sg = sigmoid_f(u);
            const float sd = sg * (1.0f - sg);
            buf[0 * HP + n] = softplus_f(u);
#pragma unroll
            for (int i = 0; i < 4; ++i) buf[(1 + i) * HP + n] = sg * g[i];
            int c = 5;
#pragma unroll
            for (int i = 0; i < 4; ++i)
#pragma unroll
                for (int j = i; j < 4; ++j, ++c)
                    buf[c * HP + n] = sd * g[i] * g[j];
            buf[15 * HP + n] = 0.0f;
        }
    }
    __syncthreads();

    // ---- layer 2: fp32 WMMA GEMM + jet activation ----
    layer_gemm_wmma(W2, bufA, bufB, lane, wave);
    __syncthreads();
    jet_activation(bufB, b2, tid);
    __syncthreads();

    // ---- layer 3 ----
    layer_gemm_wmma(W3, bufB, bufA, lane, wave);
    __syncthreads();
    jet_activation(bufA, b3, tid);
    __syncthreads();

    // ---- layer 4 (1x500): 15-channel reduction per sample (linear) ----
    {
        const int si  = tid >> 6;          // 4 samples
        const int seg = (tid >> 4) & 3;    // 4 segments
        const int ch  = tid & 15;
        const float* buf = bufA + si * SBUF + ch * HP;
        float part = 0.0f;
        for (int m = seg; m < H_DIM; m += 4) part += W4[m] * buf[m];
        red[si * 64 + seg * 16 + ch] = part;
    }
    __syncthreads();
    if (tid < 64) {
        const int si = tid >> 4, ch = tid & 15;
        float acc = 0.0f;
#pragma unroll
        for (int sgi = 0; sgi < 4; ++sgi) acc += red[si * 64 + sgi * 16 + ch];
        red[256 + si * 16 + ch] = acc;
    }
    __syncthreads();

    // ---- Euler-Lagrange solve: one thread per sample ----
    if (tid < SPB) {
        const int si = tid;
        const float* rc = red + 256 + si * 16;
        const float z2 = zsh[si * 4 + 2], z3 = zsh[si * 4 + 3];
        // channels: 1..4 = J, hess upper (0,0)(0,1)(0,2)(0,3)(1,1)(1,2)(1,3)(2,2)(2,3)(3,3)
        const float J0  = rc[1],  J1  = rc[2];
        const float h02 = rc[7],  h03 = rc[8];
        const float h12 = rc[10], h13 = rc[11];
        const float a   = rc[12], b = rc[13], c = rc[14];
        const float T0 = h02 * z2 + h12 * z3;
        const float T1 = h03 * z2 + h13 * z3;
        const float r0 = J0 - T0, r1 = J1 - T1;
        const float det = a * c - b * b;
        const float nrm = a * a + 2.0f * b * b + c * c;
        float q0, q1;
        if (fabsf(det) > 1e-10f * nrm + 1e-30f) {
            const float id = 1.0f / det;
            q0 = ( c * r0 - b * r1) * id;
            q1 = (-b * r0 + a * r1) * id;
        } else {
            const float tr = a + c;
            const float t2 = tr * tr;
            if (t2 > 1e-30f) {  // rank-1 symmetric: pinv(A) = A / tr^2
                const float isc = 1.0f / t2;
                q0 = (a * r0 + b * r1) * isc;
                q1 = (b * r0 + c * r1) * isc;
            } else { q0 = 0.0f; q1 = 0.0f; }
        }
        const int s = blockIdx.x * SPB + si;
        f_out[s * 4 + 0] = z2;
        f_out[s * 4 + 1] = z3;
        f_out[s * 4 + 2] = q0;
        f_out[s * 4 + 3] = q1;
    }
}

__global__ __launch_bounds__(256)
void axpy_kernel(float* __restrict__ out, const float* __restrict__ x,
                 const float* __restrict__ f, float alpha, int n) {
    const int i = blockIdx.x * blockDim.x + threadIdx.x;
    if (i < n) out[i] = x[i] + alpha * f[i];
}

__global__ __launch_bounds__(256)
void rk4_final_kernel(float* __restrict__ out, const float* __restrict__ x,
                      const float* __restrict__ f1, const float* __restrict__ f2,
                      const float* __restrict__ f3, const float* __restrict__ f4, int n) {
    const int i = blockIdx.x * blockDim.x + threadIdx.x;
    if (i < n)
        out[i] = x[i] + (DT_F / 6.0f) * (f1[i] + 2.0f * f2[i] + 2.0f * f3[i] + f4[i]);
}

extern "C" void kernel_launch(void* const* d_in, const int* in_sizes, int n_in,
                              void* d_out, int out_size, void* d_ws, size_t ws_size,
                              hipStream_t stream) {
    const float* x  = (const float*)d_in[0];
    const float* W1 = (const float*)d_in[1];
    const float* b1 = (const float*)d_in[2];
    const float* W2 = (const float*)d_in[3];
    const float* b2 = (const float*)d_in[4];
    const float* W3 = (const float*)d_in[5];
    const float* b3 = (const float*)d_in[6];
    const float* W4 = (const float*)d_in[7];
    const float* b4 = (const float*)d_in[8];
    float* out = (float*)d_out;

    const int nst = BATCH * ZDIM;
    float* f1  = (float*)d_ws;
    float* f2  = f1 + nst;
    float* f3  = f2 + nst;
    float* f4  = f3 + nst;
    float* tmp = f4 + nst;

    const size_t smem = (size_t)(2 * SPB * SBUF + 320 + SPB * ZDIM) * sizeof(float);
    dim3 blk(256), grd(BATCH / SPB), cgrd((nst + 255) / 256);

    lnn_f_kernel<<<grd, blk, smem, stream>>>(x,   W1, b1, W2, b2, W3, b3, W4, b4, f1);
    axpy_kernel <<<cgrd, blk, 0, stream>>>(tmp, x, f1, 0.5f * DT_F, nst);
    lnn_f_kernel<<<grd, blk, smem, stream>>>(tmp, W1, b1, W2, b2, W3, b3, W4, b4, f2);
    axpy_kernel <<<cgrd, blk, 0, stream>>>(tmp, x, f2, 0.5f * DT_F, nst);
    lnn_f_kernel<<<grd, blk, smem, stream>>>(tmp, W1, b1, W2, b2, W3, b3, W4, b4, f3);
    axpy_kernel <<<cgrd, blk, 0, stream>>>(tmp, x, f3, DT_F, nst);
    lnn_f_kernel<<<grd, blk, smem, stream>>>(tmp, W1, b1, W2, b2, W3, b3, W4, b4, f4);
    rk4_final_kernel<<<cgrd, blk, 0, stream>>>(out, x, f1, f2, f3, f4, nst);
}